// EnhancedMultiHeadAttention_24077586662051
// MI455X (gfx1250) — compile-verified
//
#include <hip/hip_runtime.h>
#include <math.h>

// Problem constants (reference: B=2, S=2048, D=512, H=8, DK=64)
#define BATCH 2
#define SLEN 2048
#define DMODEL 512
#define NHEAD 8
#define DK 64
#define NTOK (BATCH * SLEN)          // 4096
#define SCALE 0.125f                 // 1/sqrt(64)
#define DECAY 0.02f
#define LN_EPS 1e-5f

typedef __attribute__((ext_vector_type(2))) float v2f;
typedef __attribute__((ext_vector_type(8))) float v8f;
typedef __attribute__((ext_vector_type(4))) unsigned int v4u;
typedef __attribute__((ext_vector_type(8))) int v8i;
typedef __attribute__((ext_vector_type(4))) int v4i;

// LDS float offsets inside the attention kernel's dynamic smem block
#define SS_OFF    0                         // 16*2048 scores/probs
#define SPART_OFF (16 * SLEN)               // 1024 split-K partials
#define SM_OFF    (SPART_OFF + 1024)        // 16 row maxima
#define SL_OFF    (SM_OFF + 16)             // 16 row 1/sum
#define SQ_OFF    (SL_OFF + 16)             // 16*64 Q tile (TDM destination)
#define SMEM_FLOATS (SQ_OFF + 16 * DK)
#define SQ_BYTE_OFF (SQ_OFF * 4)            // dynamic-LDS base is offset 0

// FP32 WMMA: D(16x16) = A(16x4) * B(4x16) + C.  Full fp32 precision — the
// kernel is bandwidth-bound on the 268MB attn write, so fp32 matrix math is
// free relative to the roofline.
__device__ __forceinline__ v8f wmma4(v2f a, v2f b, v8f c) {
  return __builtin_amdgcn_wmma_f32_16x16x4_f32(
      /*neg_a=*/false, a, /*neg_b=*/false, b,
      /*c_mod=*/(short)0, c, /*reuse_a=*/false, /*reuse_b=*/false);
}

// ---------------------------------------------------------------------------
// Kernel 1: Y = X*W + bias (row-major).  One wave per 16x16 output tile.
// A-frag: lane(0..15)=row M, VGPR0/1 = K+2*half / +1  -> contiguous float2.
// B-frag: lane(0..15)=col N, same K striping          -> strided scalar loads.
// ---------------------------------------------------------------------------
__global__ __launch_bounds__(32) void gemm_bias_kernel(
    const float* __restrict__ X, const float* __restrict__ W,
    const float* __restrict__ bias, float* __restrict__ Y,
    int M, int K, int N) {
  const int m0 = blockIdx.x * 16;
  const int n0 = blockIdx.y * 16;
  const int lane = threadIdx.x;
  const int half = lane >> 4;
  const int nl = lane & 15;

  v8f c = {};
  const float* arow = X + (size_t)(m0 + nl) * K;
  for (int k0 = 0; k0 < K; k0 += 4) {
    const int kk = k0 + 2 * half;
    v2f a = *(const v2f*)(arow + kk);              // global_load_b64
    v2f b;
    b.x = W[(size_t)kk * N + n0 + nl];
    b.y = W[(size_t)(kk + 1) * N + n0 + nl];
    c = wmma4(a, b, c);
  }
#pragma unroll
  for (int j = 0; j < 8; ++j) {
    const int m = m0 + j + 8 * half;               // C/D layout: VGPRj -> M=j+8*half
    const int n = n0 + nl;
    Y[(size_t)m * N + n] = c[j] + bias[n];
  }
}

// ---------------------------------------------------------------------------
// Kernel 2: fused attention for one (b,h, 16-row q tile).
// 8 waves / 256 threads.  The whole 16x2048 score strip lives in LDS
// (CDNA5: 320KB per WGP -> ~136KB dynamic LDS here, 2 WGs/WGP).
//   phase 0: TDM async-loads the 16x64 Q tile into LDS (tensor_load_to_lds)
//   phase 1: score tiles (QK^T*scale + tril exp bias) via WMMA -> LDS
//   phase 2: two-pass softmax in LDS (shfl-16 reductions), coalesced write of
//            normalized probs to the attn output (single HBM pass, 268MB total)
//   phase 3: ctx = P*V via WMMA, split-K over 2 wave groups + LDS reduce
// ---------------------------------------------------------------------------
__global__ __launch_bounds__(256) void attn_kernel(
    const float* __restrict__ q, const float* __restrict__ k,
    const float* __restrict__ v, float* __restrict__ attn,
    float* __restrict__ ctx) {
  extern __shared__ float smem[];
  float* sS = smem + SS_OFF;
  float* sPart = smem + SPART_OFF;
  float* sM = smem + SM_OFF;
  float* sL = smem + SL_OFF;
  float* sQ = smem + SQ_OFF;

  const int qt = blockIdx.x;            // 0..127 q tile
  const int bh = blockIdx.y;            // 0..15  (b*H + h)
  const int b = bh >> 3;
  const int h = bh & 7;
  const int tid = threadIdx.x;
  const int wave = tid >> 5;
  const int lane = tid & 31;
  const int half = lane >> 4;
  const int nl = lane & 15;
  const int hoff = h * DK;
  const int qrow0 = qt * 16;            // q index within sequence

  const float* qbase = q + (size_t)b * SLEN * DMODEL;
  const float* kbase = k + (size_t)b * SLEN * DMODEL;
  const float* vbase = v + (size_t)b * SLEN * DMODEL;

  // ---- phase 0: TDM load of the Q tile (16 rows x 256B, row stride 2KB) ----
  // D# built per ISA §8: group0={count,lds_addr,global_addr,type=2},
  // group1={data_size=4B, tensor_dim0/1, tile_dim0/1, dim0_stride}, 2D tensor
  // so remaining groups are zero.  Values are block-uniform -> SGPRs.
  if (wave == 0) {
    const unsigned long long ga =
        (unsigned long long)(uintptr_t)(qbase + (size_t)qrow0 * DMODEL + hoff);
    v4u g0;
    g0[0] = 1u;                                            // count=1, user D#
    g0[1] = (unsigned)SQ_BYTE_OFF;                         // lds_addr
    g0[2] = (unsigned)(ga & 0xFFFFFFFFull);                // global_addr lo
    g0[3] = (unsigned)((ga >> 32) & 0x01FFFFFFull) | (2u << 30);  // hi | type=2
    v8i g1;
    g1[0] = (int)(2u << 16);                               // data_size = 4 bytes
    g1[1] = (int)(((unsigned)DMODEL) << 16);               // tensor_dim0 = 512
    g1[2] = (int)(((unsigned)SLEN) << 16);                 // tensor_dim1 = 2048
    g1[3] = (int)(((unsigned)DK) << 16);                   // tile_dim0 = 64
    g1[4] = 16;                                            // tile_dim1 = 16
    g1[5] = DMODEL;                                        // tensor_dim0_stride
    g1[6] = 0;
    g1[7] = 0;
    v4i gz4 = {0, 0, 0, 0};
    v8i gz8 = {0, 0, 0, 0, 0, 0, 0, 0};
    // 6-arg variant on this toolchain (clang-23 lane per CDNA5_HIP.md)
    __builtin_amdgcn_tensor_load_to_lds(g0, g1, gz4, gz4, gz8, 0);
    __builtin_amdgcn_s_wait_tensorcnt(0);
  }
  __syncthreads();

  // Q fragments for all 16 K-steps from LDS (reused across 128 key tiles).
  v2f qa[16];
#pragma unroll
  for (int s = 0; s < 16; ++s)
    qa[s] = *(const v2f*)(sQ + nl * DK + s * 4 + 2 * half);   // ds_load_b64

  // ---- phase 1: scores -> LDS ----
  for (int kt = wave; kt < SLEN / 16; kt += 8) {
    const int key0 = kt * 16;
    const float* krow = kbase + (size_t)(key0 + nl) * DMODEL + hoff;
    if (kt + 8 < SLEN / 16)   // hide L2 latency for next tile's strided rows
      __builtin_prefetch(kbase + (size_t)(key0 + 128 + nl) * DMODEL + hoff, 0, 0);
    v8f c = {};
#pragma unroll
    for (int s = 0; s < 16; ++s) {
      v2f bf = *(const v2f*)(krow + s * 4 + 2 * half);   // B[k][n]=K[key n][k]
      c = wmma4(qa[s], bf, c);
    }
#pragma unroll
    for (int j = 0; j < 8; ++j) {
      const int m = j + 8 * half;
      const int qi = qrow0 + m;
      const int kj = key0 + nl;
      float sc = c[j] * SCALE;
      if (kj <= qi) sc += __expf(-DECAY * (float)(qi - kj));  // additive tril bias
      sS[m * SLEN + kj] = sc;
    }
  }
  __syncthreads();

  // ---- phase 2a: per-row max & sum (16 threads per row, shfl width 16) ----
  {
    const int row = tid >> 4;
    const int c0 = tid & 15;
    float mx = -3.4e38f;
    for (int cc = c0; cc < SLEN; cc += 16) mx = fmaxf(mx, sS[row * SLEN + cc]);
#pragma unroll
    for (int off = 8; off >= 1; off >>= 1) mx = fmaxf(mx, __shfl_xor(mx, off, 16));
    float sum = 0.f;
    for (int cc = c0; cc < SLEN; cc += 16) sum += __expf(sS[row * SLEN + cc] - mx);
#pragma unroll
    for (int off = 8; off >= 1; off >>= 1) sum += __shfl_xor(sum, off, 16);
    if (c0 == 0) { sM[row] = mx; sL[row] = 1.0f / sum; }
  }
  __syncthreads();

  // ---- phase 2b: normalize in LDS + single coalesced HBM write of attn ----
  {
    float* attn_base = attn + ((size_t)bh * SLEN + qrow0) * SLEN;
    for (int idx = tid; idx < 16 * SLEN; idx += 256) {
      const int row = idx >> 11;
      const float p = __expf(sS[idx] - sM[row]) * sL[row];
      sS[idx] = p;
      attn_base[idx] = p;                 // sS layout == attn row-major layout
    }
  }
  __syncthreads();

  // ---- phase 3: ctx tile = P(16x2048) * V(2048x64), split-K over 2 groups ----
  {
    const int ct = wave & 3;              // DK column tile 0..3
    const int khalf = wave >> 2;          // key range half
    const float* vcol = vbase + hoff + ct * 16;
    v8f c = {};
    const int kbeg = khalf * (SLEN / 2);
    for (int k0 = kbeg; k0 < kbeg + SLEN / 2; k0 += 4) {
      const int kk = k0 + 2 * half;
      v2f a = *(const v2f*)(sS + nl * SLEN + kk);        // ds_load_b64
      v2f bf;
      bf.x = vcol[(size_t)kk * DMODEL + nl];
      bf.y = vcol[(size_t)(kk + 1) * DMODEL + nl];
      c = wmma4(a, bf, c);
    }
    if (khalf == 1) {
#pragma unroll
      for (int j = 0; j < 8; ++j) sPart[ct * 256 + j * 32 + lane] = c[j];
    }
    __syncthreads();
    if (khalf == 0) {
      float* out = ctx + ((size_t)b * SLEN + qrow0) * DMODEL + hoff + ct * 16;
#pragma unroll
      for (int j = 0; j < 8; ++j) {
        const int m = j + 8 * half;
        out[(size_t)m * DMODEL + nl] = c[j] + sPart[ct * 256 + j * 32 + lane];
      }
    }
  }
}

// ---------------------------------------------------------------------------
// Kernel 3: per 16-token tile: out = ctx*wo + bo (WMMA), gate MLP (VALU,
// ~1 GF total), gated residual, LayerNorm.  8 waves / 256 threads.
// ---------------------------------------------------------------------------
__global__ __launch_bounds__(256) void out_proj_ln_kernel(
    const float* __restrict__ ctx, const float* __restrict__ wo,
    const float* __restrict__ bo, const float* __restrict__ x,
    const float* __restrict__ g1w, const float* __restrict__ g1b,
    const float* __restrict__ g2w, const float* __restrict__ g2b,
    const float* __restrict__ lng, const float* __restrict__ lnb,
    float* __restrict__ y) {
  __shared__ float sO[16 * DMODEL];     // 32KB
  __shared__ float sPartG[256];
  __shared__ float sGate[16];
  __shared__ float sMu[16], sRstd[16];

  const int t0 = blockIdx.x * 16;       // token tile base
  const int tid = threadIdx.x;
  const int wave = tid >> 5;
  const int lane = tid & 31;
  const int half = lane >> 4;
  const int nl = lane & 15;

  // gate MLP partials: thread -> (token = tid&15, 16-wide hidden chunk)
  {
    const int tok = tid & 15;
    const int jc = tid >> 4;
    const float* xr = x + (size_t)(t0 + tok) * DMODEL;
    float partial = 0.f;
    for (int j = 0; j < 16; ++j) {
      const int jj = jc * 16 + j;
      float hsum = g1b[jj];
      for (int kk = 0; kk < DMODEL; ++kk)
        hsum = fmaf(xr[kk], g1w[(size_t)kk * (DMODEL / 2) + jj], hsum);
      partial = fmaf(fmaxf(hsum, 0.f), g2w[jj], partial);
    }
    sPartG[tid] = partial;
  }
  __syncthreads();
  if (tid < 16) {
    float s = g2b[0];
    for (int jc = 0; jc < 16; ++jc) s += sPartG[jc * 16 + tid];
    sGate[tid] = 1.f / (1.f + __expf(-s));
  }

  // output projection: each wave owns 4 of the 32 16-wide column tiles
  for (int ci = 0; ci < 4; ++ci) {
    const int n0 = (wave * 4 + ci) * 16;
    const float* arow = ctx + (size_t)(t0 + nl) * DMODEL;
    v8f c = {};
    for (int k0 = 0; k0 < DMODEL; k0 += 4) {
      const int kk = k0 + 2 * half;
      v2f a = *(const v2f*)(arow + kk);
      v2f bf;
      bf.x = wo[(size_t)kk * DMODEL + n0 + nl];
      bf.y = wo[(size_t)(kk + 1) * DMODEL + n0 + nl];
      c = wmma4(a, bf, c);
    }
#pragma unroll
    for (int j = 0; j < 8; ++j)
      sO[(j + 8 * half) * DMODEL + n0 + nl] = c[j] + bo[n0 + nl];
  }
  __syncthreads();

  // gated residual (in LDS)
  for (int idx = tid; idx < 16 * DMODEL; idx += 256) {
    const int row = idx >> 9;
    const int col = idx & (DMODEL - 1);
    const float g = sGate[row];
    sO[idx] = sO[idx] * g + x[(size_t)(t0 + row) * DMODEL + col] * (1.f - g);
  }
  __syncthreads();

  // LayerNorm stats: 16 threads per row, shfl-16 reduce
  {
    const int row = tid >> 4;
    const int c0 = tid & 15;
    float s = 0.f, s2 = 0.f;
    for (int cc = c0; cc < DMODEL; cc += 16) {
      const float vv = sO[row * DMODEL + cc];
      s += vv; s2 = fmaf(vv, vv, s2);
    }
#pragma unroll
    for (int off = 8; off >= 1; off >>= 1) {
      s += __shfl_xor(s, off, 16);
      s2 += __shfl_xor(s2, off, 16);
    }
    if (c0 == 0) {
      const float mu = s * (1.f / DMODEL);
      const float var = s2 * (1.f / DMODEL) - mu * mu;  // population var (jnp.var)
      sMu[row] = mu;
      sRstd[row] = rsqrtf(var + LN_EPS);
    }
  }
  __syncthreads();

  for (int idx = tid; idx < 16 * DMODEL; idx += 256) {
    const int row = idx >> 9;
    const int col = idx & (DMODEL - 1);
    y[(size_t)(t0 + row) * DMODEL + col] =
        (sO[idx] - sMu[row]) * sRstd[row] * lng[col] + lnb[col];
  }
}

// ---------------------------------------------------------------------------
extern "C" void kernel_launch(void* const* d_in, const int* in_sizes, int n_in,
                              void* d_out, int out_size, void* d_ws, size_t ws_size,
                              hipStream_t stream) {
  const float* x   = (const float*)d_in[0];
  const float* wq  = (const float*)d_in[1];
  const float* bq  = (const float*)d_in[2];
  const float* wk  = (const float*)d_in[3];
  const float* bk  = (const float*)d_in[4];
  const float* wv  = (const float*)d_in[5];
  const float* bv  = (const float*)d_in[6];
  const float* wo  = (const float*)d_in[7];
  const float* bo  = (const float*)d_in[8];
  const float* g1w = (const float*)d_in[9];
  const float* g1b = (const float*)d_in[10];
  const float* g2w = (const float*)d_in[11];
  const float* g2b = (const float*)d_in[12];
  const float* lng = (const float*)d_in[13];
  const float* lnb = (const float*)d_in[14];

  float* y    = (float*)d_out;                        // (B,S,D)
  float* attn = y + (size_t)NTOK * DMODEL;            // (B,H,S,S)

  // workspace: q,k,v,ctx each NTOK*DMODEL floats (32MB total)
  float* q   = (float*)d_ws;
  float* k   = q + (size_t)NTOK * DMODEL;
  float* v   = k + (size_t)NTOK * DMODEL;
  float* ctx = v + (size_t)NTOK * DMODEL;

  // QKV projections
  dim3 gproj(NTOK / 16, DMODEL / 16);
  gemm_bias_kernel<<<gproj, 32, 0, stream>>>(x, wq, bq, q, NTOK, DMODEL, DMODEL);
  gemm_bias_kernel<<<gproj, 32, 0, stream>>>(x, wk, bk, k, NTOK, DMODEL, DMODEL);
  gemm_bias_kernel<<<gproj, 32, 0, stream>>>(x, wv, bv, v, NTOK, DMODEL, DMODEL);

  // fused attention: ~136KB dynamic LDS (fits 2 workgroups per 320KB WGP)
  const size_t lds_bytes = (size_t)SMEM_FLOATS * sizeof(float);
  (void)hipFuncSetAttribute(reinterpret_cast<const void*>(attn_kernel),
                            hipFuncAttributeMaxDynamicSharedMemorySize,
                            (int)lds_bytes);
  attn_kernel<<<dim3(SLEN / 16, BATCH * NHEAD), 256, lds_bytes, stream>>>(
      q, k, v, attn, ctx);

  // output projection + gate + residual + LN
  out_proj_ln_kernel<<<NTOK / 16, 256, 0, stream>>>(
      ctx, wo, bo, x, g1w, g1b, g2w, g2b, lng, lnb, y);
}